// GPT_21887153340720
// MI455X (gfx1250) — compile-verified
//
#include <hip/hip_runtime.h>
#include <hip/hip_bf16.h>

// ---------------------------------------------------------------------------
// GPT forward (fp32) for gfx1250: V_WMMA_F32_16X16X4_F32 for all GEMMs,
// GLOBAL_LOAD_ASYNC_TO_LDS_B128 double-buffered tile staging in the GEMM.
// B=2, T=1024, D=1024, H=16, HD=64, L=8, V=32000.
// ---------------------------------------------------------------------------

typedef float v2f __attribute__((ext_vector_type(2)));
typedef float v8f __attribute__((ext_vector_type(8)));

#define NTOK 2048      // B*T
#define DMODEL 1024
#define NHEAD 16
#define HDIM 64
#define NLAYER 8
#define VOCAB 32000
#define DFF 4096

// -------------------- CDNA5 async global->LDS helpers ----------------------
// lds_off: byte offset within the workgroup's LDS allocation (low 32 bits of
// the flat pointer to __shared__ memory; ISA 10.2 truncates flat LDS
// addresses to the 32-bit LDS offset). 16-byte aligned for b128.
__device__ __forceinline__ void async_copy_b128(unsigned lds_off,
                                                const float* gaddr) {
  asm volatile("global_load_async_to_lds_b128 %0, %1, off"
               :: "v"(lds_off), "v"(gaddr)
               : "memory");
}
__device__ __forceinline__ void async_wait_all() {
  asm volatile("s_wait_asynccnt 0x0" ::: "memory");
}

// ------------------------------- embedding ---------------------------------
__global__ void embed_kernel(const int* __restrict__ tok,
                             const float* __restrict__ wte,
                             const float* __restrict__ wpe,
                             float* __restrict__ x) {
  int row = blockIdx.x;            // 0..2047  (b*T + t)
  int t = row & 1023;
  int tk = tok[row];
  const float* te = wte + (size_t)tk * DMODEL;
  const float* pe = wpe + (size_t)t * DMODEL;
  float* xr = x + (size_t)row * DMODEL;
  for (int d = threadIdx.x; d < DMODEL; d += 256) xr[d] = te[d] + pe[d];
}

// ------------------------------- layernorm ---------------------------------
__global__ void ln_kernel(const float* __restrict__ x,
                          const float* __restrict__ g,
                          const float* __restrict__ bta,
                          float* __restrict__ out) {
  __shared__ float red0[8];
  __shared__ float red1[8];
  int row = blockIdx.x;
  const float* xr = x + (size_t)row * DMODEL;
  int tid = threadIdx.x, wave = tid >> 5, lane = tid & 31;
  float s = 0.f, ss = 0.f;
  float vals[4];
  for (int i = 0; i < 4; ++i) {
    float t = xr[tid + i * 256];
    vals[i] = t;
    s += t; ss += t * t;
  }
  for (int off = 16; off; off >>= 1) {
    s  += __shfl_xor(s,  off, 32);
    ss += __shfl_xor(ss, off, 32);
  }
  if (lane == 0) { red0[wave] = s; red1[wave] = ss; }
  __syncthreads();
  if (wave == 0) {
    float a = (lane < 8) ? red0[lane] : 0.f;
    float b = (lane < 8) ? red1[lane] : 0.f;
    for (int off = 4; off; off >>= 1) {
      a += __shfl_xor(a, off, 32);
      b += __shfl_xor(b, off, 32);
    }
    if (lane == 0) { red0[0] = a; red1[0] = b; }
  }
  __syncthreads();
  float mean = red0[0] * (1.0f / DMODEL);
  float var  = red1[0] * (1.0f / DMODEL) - mean * mean;
  float inv  = rsqrtf(var + 1e-5f);
  float* orow = out + (size_t)row * DMODEL;
  for (int i = 0; i < 4; ++i) {
    int d = tid + i * 256;
    orow[d] = (vals[i] - mean) * inv * g[d] + bta[d];
  }
}

// ----------------------- repack (H,D,HD) -> (D, H*HD) ----------------------
__global__ void repack_qkv_kernel(const float* __restrict__ w,
                                  float* __restrict__ out) {
  int idx = blockIdx.x * 256 + threadIdx.x;   // over 1M elements
  if (idx < DMODEL * DMODEL) {
    int d = idx >> 10;
    int n = idx & 1023;
    int h = n >> 6, k = n & 63;
    out[idx] = w[((size_t)h * DMODEL + d) * HDIM + k];
  }
}

// --------------------------- generic WMMA GEMM -----------------------------
// out[M,N] = act( A[M,K] @ W[K,N] + bias[N] (+ resid[M,N]) )
// 64x64 block tile, K in chunks of 32 through double-buffered LDS filled by
// global_load_async_to_lds_b128; 8 waves x two 16x16 v_wmma accumulators.
#define BM 64
#define BN 64
#define BK 32
#define ASTR 36   // A LDS row stride (floats): 144B, 16B-aligned, non-pow2
#define BSTR 68   // B LDS row stride (floats): 272B, 16B-aligned, non-pow2
__global__ __launch_bounds__(256)
void gemm_kernel(const float* __restrict__ A, const float* __restrict__ W,
                 const float* __restrict__ bias, const float* __restrict__ resid,
                 float* __restrict__ out, int M, int N, int K, int relu) {
  __shared__ __align__(16) float As[2][BM * ASTR];
  __shared__ __align__(16) float Bs[2][BK * BSTR];
  int tid = threadIdx.x;
  int wave = tid >> 5, lane = tid & 31;
  int m0 = blockIdx.y * BM;
  int n0 = blockIdx.x * BN;
  int koff = (lane >> 4) * 2;      // A/B K sub-offset per half-wave (ISA layout)

  // async-copy coordinates: 512 16B chunks per tile, 2 per thread per matrix
  int am1 = tid >> 3,          ac1 = (tid & 7) * 4;
  int am2 = (tid + 256) >> 3,  ac2 = ((tid + 256) & 7) * 4;
  int bk1 = tid >> 4,          bc1 = (tid & 15) * 4;
  int bk2 = (tid + 256) >> 4,  bc2 = ((tid + 256) & 15) * 4;
  unsigned asb[2] = { (unsigned)(uintptr_t)&As[0][0],
                      (unsigned)(uintptr_t)&As[1][0] };
  unsigned bsb[2] = { (unsigned)(uintptr_t)&Bs[0][0],
                      (unsigned)(uintptr_t)&Bs[1][0] };

#define ISSUE_TILE(buf, kk)                                                   \
  do {                                                                        \
    async_copy_b128(asb[buf] + (unsigned)(am1 * ASTR + ac1) * 4,              \
                    A + (size_t)(m0 + am1) * K + (kk) + ac1);                 \
    async_copy_b128(asb[buf] + (unsigned)(am2 * ASTR + ac2) * 4,              \
                    A + (size_t)(m0 + am2) * K + (kk) + ac2);                 \
    async_copy_b128(bsb[buf] + (unsigned)(bk1 * BSTR + bc1) * 4,              \
                    W + (size_t)((kk) + bk1) * N + n0 + bc1);                 \
    async_copy_b128(bsb[buf] + (unsigned)(bk2 * BSTR + bc2) * 4,              \
                    W + (size_t)((kk) + bk2) * N + n0 + bc2);                 \
  } while (0)

  v8f c0 = {}, c1 = {};
  ISSUE_TILE(0, 0);
  for (int kk = 0; kk < K; kk += BK) {
    int buf = (kk >> 5) & 1;
    async_wait_all();          // my tile copies have landed in LDS
    __syncthreads();           // everyone's have; prior reads of buf^1 done
    if (kk + BK < K) ISSUE_TILE(buf ^ 1, kk + BK);
    const float* Asb = &As[buf][0];
    const float* Bsb = &Bs[buf][0];
    for (int t = 0; t < 2; ++t) {
      int tile = wave * 2 + t;
      int am = (tile >> 2) * 16 + (lane & 15);
      int bn = (tile & 3) * 16 + (lane & 15);
      v8f acc = t ? c1 : c0;
      for (int k4 = 0; k4 < BK; k4 += 4) {
        v2f a, b;
        a.x = Asb[am * ASTR + k4 + koff];
        a.y = Asb[am * ASTR + k4 + koff + 1];
        b.x = Bsb[(k4 + koff) * BSTR + bn];
        b.y = Bsb[(k4 + koff + 1) * BSTR + bn];
        acc = __builtin_amdgcn_wmma_f32_16x16x4_f32(false, a, false, b,
                                                    (short)0, acc, false, false);
      }
      if (t) c1 = acc; else c0 = acc;
    }
  }
#undef ISSUE_TILE
  for (int t = 0; t < 2; ++t) {
    int tile = wave * 2 + t;
    int tm = (tile >> 2) * 16, tn = (tile & 3) * 16;
    int n = n0 + tn + (lane & 15);
    int mbase = m0 + tm + ((lane >> 4) ? 8 : 0);
    float bv = bias ? bias[n] : 0.0f;
    v8f acc = t ? c1 : c0;
    for (int r = 0; r < 8; ++r) {
      size_t idx = (size_t)(mbase + r) * N + n;
      float val = acc[r] + bv;
      if (resid) val += resid[idx];
      if (relu) val = fmaxf(val, 0.0f);
      out[idx] = val;
    }
  }
}

// ------------------------------ attention ----------------------------------
// One block = 32 query rows of one (b,h). The full 32x1024 fp32 score tile
// lives in LDS (~132 KB of the WGP's 320 KB). Faithful to the reference:
// masked (upper-triangle) scores are set to 0 BEFORE a full-row softmax,
// so masked positions contribute exp(0-max)/Z to the P@V product.
#define AROWS 32
#define SSTR  1032      // 1024 + 8 pad to break LDS bank conflicts
#define QSTR  68
__global__ __launch_bounds__(256)
void attn_kernel(const float* __restrict__ q, const float* __restrict__ k,
                 const float* __restrict__ v, float* __restrict__ o) {
  extern __shared__ float smem[];
  float* S  = smem;                 // AROWS * SSTR
  float* Qs = smem + AROWS * SSTR;  // AROWS * QSTR
  int tid = threadIdx.x, wave = tid >> 5, lane = tid & 31;
  int t0 = blockIdx.x * AROWS;
  int bh = blockIdx.y;
  int b = bh >> 4, h = bh & 15;
  // layout of q/k/v/o: [b, t, h, hd] flat -> base + t*1024 + hd
  size_t base = (size_t)b * 1024 * DMODEL + (size_t)h * HDIM;
  int koff = (lane >> 4) * 2;
  int nl = lane & 15;

  for (int e = tid; e < AROWS * HDIM; e += 256) {
    int m = e >> 6, d = e & 63;
    Qs[m * QSTR + d] = q[base + (size_t)(t0 + m) * DMODEL + d];
  }
  __syncthreads();

  // S = Q @ K^T  (masked): 64 column tiles of 16 keys, strided over waves.
  for (int ct = wave; ct < 64; ct += 8) {
    int s0 = ct * 16;
    const float* kcol = k + base + (size_t)(s0 + nl) * DMODEL;
    v2f bf[16];
    for (int k4 = 0; k4 < 16; ++k4) {
      bf[k4].x = kcol[k4 * 4 + koff];
      bf[k4].y = kcol[k4 * 4 + koff + 1];
    }
    for (int rt = 0; rt < 2; ++rt) {
      int am = rt * 16 + nl;
      v8f c = {};
      for (int k4 = 0; k4 < 16; ++k4) {
        v2f a;
        a.x = Qs[am * QSTR + k4 * 4 + koff];
        a.y = Qs[am * QSTR + k4 * 4 + koff + 1];
        c = __builtin_amdgcn_wmma_f32_16x16x4_f32(false, a, false, bf[k4],
                                                  (short)0, c, false, false);
      }
      int scol = s0 + nl;
      int mrow0 = rt * 16 + ((lane >> 4) ? 8 : 0);
      for (int r = 0; r < 8; ++r) {
        int mrow = mrow0 + r;
        int tq = t0 + mrow;
        S[mrow * SSTR + scol] = (scol <= tq) ? c[r] : 0.0f;   // zero, not -inf
      }
    }
  }
  __syncthreads();

  // Full-row softmax over all 1024 entries.
  for (int r = wave; r < AROWS; r += 8) {
    float* row = S + r * SSTR;
    float mx = -3.402823466e38f;
    for (int i = lane; i < 1024; i += 32) mx = fmaxf(mx, row[i]);
    for (int off = 16; off; off >>= 1) mx = fmaxf(mx, __shfl_xor(mx, off, 32));
    float sum = 0.f;
    for (int i = lane; i < 1024; i += 32) {
      float e = __expf(row[i] - mx);
      row[i] = e;
      sum += e;
    }
    for (int off = 16; off; off >>= 1) sum += __shfl_xor(sum, off, 32);
    float inv = 1.0f / sum;
    for (int i = lane; i < 1024; i += 32) row[i] *= inv;
  }
  __syncthreads();

  // O = P @ V : 2x4 grid of 16x16 tiles, one per wave; K-dim = 1024.
  {
    int rt = wave >> 2, cn = wave & 3;
    int am = rt * 16 + nl;
    int bn = cn * 16 + nl;
    v8f c = {};
    for (int k4 = 0; k4 < 256; ++k4) {
      int kb = k4 * 4 + koff;
      v2f a, bf;
      a.x = S[am * SSTR + kb];
      a.y = S[am * SSTR + kb + 1];
      const float* vrow = v + base + (size_t)kb * DMODEL + bn;
      bf.x = vrow[0];
      bf.y = vrow[DMODEL];
      c = __builtin_amdgcn_wmma_f32_16x16x4_f32(false, a, false, bf,
                                                (short)0, c, false, false);
    }
    int mrow0 = rt * 16 + ((lane >> 4) ? 8 : 0);
    int col = cn * 16 + nl;
    for (int r = 0; r < 8; ++r) {
      int tq = t0 + mrow0 + r;
      o[base + (size_t)tq * DMODEL + col] = c[r];
    }
  }
}

// ------------------------------- driver ------------------------------------
extern "C" void kernel_launch(void* const* d_in, const int* in_sizes, int n_in,
                              void* d_out, int out_size, void* d_ws, size_t ws_size,
                              hipStream_t stream) {
  (void)in_sizes; (void)n_in; (void)out_size; (void)ws_size;
  const int*   tok = (const int*)d_in[0];
  const float* wte = (const float*)d_in[1];
  const float* wpe = (const float*)d_in[2];
  // blocks: d_in[3 + 16*l + j], j: 0 ln1_g,1 ln1_b,2 wq,3 bq,4 wk,5 bk,
  //         6 wv,7 bv,8 wproj,9 bproj,10 ln2_g,11 ln2_b,12 w1,13 b1,14 w2,15 b2
  const float* ln_g = (const float*)d_in[3 + 16 * NLAYER + 0];
  const float* ln_b = (const float*)d_in[3 + 16 * NLAYER + 1];
  const float* wout = (const float*)d_in[3 + 16 * NLAYER + 2];
  const float* bout = (const float*)d_in[3 + 16 * NLAYER + 3];

  float* ws = (float*)d_ws;
  float* x   = ws;                      // [2048,1024]
  float* hb  = x   + (size_t)NTOK * DMODEL;
  float* qb  = hb  + (size_t)NTOK * DMODEL;
  float* kb  = qb  + (size_t)NTOK * DMODEL;
  float* vb  = kb  + (size_t)NTOK * DMODEL;
  float* ob  = vb  + (size_t)NTOK * DMODEL;
  float* mh  = ob  + (size_t)NTOK * DMODEL;   // [2048,4096]
  float* wqr = mh  + (size_t)NTOK * DFF;
  float* wkr = wqr + (size_t)DMODEL * DMODEL;
  float* wvr = wkr + (size_t)DMODEL * DMODEL;

  float* logits = (float*)d_out;                       // [2048, 32000]
  float* xout   = logits + (size_t)NTOK * VOCAB;       // [2048, 1024]

  dim3 blk(256);
  dim3 gD(DMODEL / BN, NTOK / BM);     // N=1024 GEMMs
  dim3 gF(DFF / BN, NTOK / BM);        // N=4096 GEMM
  dim3 gV(VOCAB / BN, NTOK / BM);      // N=32000 GEMM
  dim3 gA(1024 / AROWS, 2 * NHEAD);    // attention
  size_t attn_lds = (size_t)(AROWS * SSTR + AROWS * QSTR) * sizeof(float);

  embed_kernel<<<NTOK, blk, 0, stream>>>(tok, wte, wpe, x);

  for (int l = 0; l < NLAYER; ++l) {
    const float* const* p = (const float* const*)(d_in + 3 + 16 * l);
    ln_kernel<<<NTOK, blk, 0, stream>>>(x, p[0], p[1], hb);
    repack_qkv_kernel<<<4096, blk, 0, stream>>>(p[2], wqr);
    repack_qkv_kernel<<<4096, blk, 0, stream>>>(p[4], wkr);
    repack_qkv_kernel<<<4096, blk, 0, stream>>>(p[6], wvr);
    gemm_kernel<<<gD, blk, 0, stream>>>(hb, wqr, p[3], nullptr, qb,
                                        NTOK, DMODEL, DMODEL, 0);
    gemm_kernel<<<gD, blk, 0, stream>>>(hb, wkr, p[5], nullptr, kb,
                                        NTOK, DMODEL, DMODEL, 0);
    gemm_kernel<<<gD, blk, 0, stream>>>(hb, wvr, p[7], nullptr, vb,
                                        NTOK, DMODEL, DMODEL, 0);
    attn_kernel<<<gA, blk, attn_lds, stream>>>(qb, kb, vb, ob);
    gemm_kernel<<<gD, blk, 0, stream>>>(ob, p[8], p[9], x, x,
                                        NTOK, DMODEL, DMODEL, 0);
    ln_kernel<<<NTOK, blk, 0, stream>>>(x, p[10], p[11], hb);
    gemm_kernel<<<gF, blk, 0, stream>>>(hb, p[12], p[13], nullptr, mh,
                                        NTOK, DFF, DMODEL, 1);
    gemm_kernel<<<gD, blk, 0, stream>>>(mh, p[14], p[15], x, x,
                                        NTOK, DMODEL, DFF, 0);
  }

  ln_kernel<<<NTOK, blk, 0, stream>>>(x, ln_g, ln_b, xout);
  gemm_kernel<<<gV, blk, 0, stream>>>(xout, wout, bout, nullptr, logits,
                                      NTOK, VOCAB, DMODEL, 0);
}